// SoftmaxCategoricalHead_70265664963187
// MI455X (gfx1250) — compile-verified
//
#include <hip/hip_runtime.h>
#include <stdint.h>

// ---- problem constants (from reference) ----
#define VOC   50257
#define BB    8
#define SS    64
#define VW    1571          // ceil(VOC/32)
#define CHV   2048          // v-span per block (chunk)
#define NCH   25            // ceil(VOC/CHV) -> 25*2048 = 51200 >= 50257
#define TT    512           // tile v-extent staged in LDS
#define NT    4             // tiles per chunk (NT*TT == CHV)
#define LOG2E 1.4426950408889634f
#define INV_TEMP (1.0f/0.6f)
#define PEN      1.2f
#define INV_PEN  (1.0f/1.2f)

// ---- CDNA5 async global->LDS helpers (gfx1250) ----
__device__ __forceinline__ void async_ld_b32(unsigned lds_off, const float* gaddr) {
    // GV form: vdst = LDS byte offset, v[addr] = 64-bit global VA, saddr = off
    asm volatile("global_load_async_to_lds_b32 %0, %1, off"
                 :: "v"(lds_off), "v"(gaddr) : "memory");
}
__device__ __forceinline__ void wait_async_le16() {
    asm volatile("s_wait_asynccnt 0x10" ::: "memory");
}
__device__ __forceinline__ void wait_async_0() {
    asm volatile("s_wait_asynccnt 0x0" ::: "memory");
}

// stage one tile: 8 batch rows x TT floats for sequence s starting at v=vb
__device__ __forceinline__ void issue_tile(const float* __restrict__ logits,
                                           int s, int vb, float* buf) {
    const int t = threadIdx.x;
    const unsigned lbase = (unsigned)(uintptr_t)buf;   // low 32 bits = LDS offset
#pragma unroll
    for (int i = 0; i < (BB * TT) / 256; ++i) {        // 16 issues/thread
        int j   = i * 256 + t;                         // [0, 4096)
        int row = j >> 9;                              // batch index (TT=512/row)
        int col = j & (TT - 1);
        int v   = vb + col;
        int gv  = (v < VOC) ? v : (VOC - 1);           // clamp: no OOB, masked later
        const float* g = logits + ((size_t)row * SS + s) * (size_t)VOC + gv;
        unsigned l = lbase + (unsigned)(row * TT + col) * 4u;
        async_ld_b32(l, g);
    }
}

__device__ __forceinline__ void combine(float& m, float& s, float om, float os) {
    float nm = fmaxf(m, om);
    float ns = s * exp2f((m - nm) * LOG2E) + os * exp2f((om - nm) * LOG2E);
    if (nm == -INFINITY) ns = 0.f;                     // both empty -> avoid NaN
    m = nm; s = ns;
}

// ---- kernel 1: zero mask words ----
__global__ void k_maskzero(unsigned* __restrict__ mask) {
    int i = blockIdx.x * blockDim.x + threadIdx.x;
    if (i < SS * VW) mask[i] = 0u;
}

// ---- kernel 2: scatter prev_tokens into bitmask (set semantics) ----
__global__ void k_maskfill(const int* __restrict__ prev, unsigned* __restrict__ mask, int P) {
    int i = blockIdx.x * blockDim.x + threadIdx.x;
    if (i < SS * P) {
        int s   = i / P;
        int tok = prev[i];
        if (tok >= 0 && tok < VOC)
            atomicOr(&mask[s * VW + (tok >> 5)], 1u << (tok & 31));
    }
}

// ---- kernel 3: per-(s,chunk) online-softmax partial (max,sum) for all 8 batches ----
__global__ __launch_bounds__(256) void k_reduce(const float* __restrict__ logits,
                                                const unsigned* __restrict__ mask,
                                                float* __restrict__ part) {
    __shared__ __align__(16) float tile[2][BB][TT];    // 32 KB double buffer
    __shared__ unsigned mw[CHV / 32];                  // 64 words
    __shared__ float redm[8][BB], reds[8][BB];
    const int t  = threadIdx.x;
    const int s  = blockIdx.x;
    const int c  = blockIdx.y;
    const int v0 = c * CHV;

    if (t < CHV / 32) {
        int wi = v0 / 32 + t;
        mw[t] = (wi < VW) ? mask[s * VW + wi] : 0u;
    }

    float m[BB], sum[BB];
#pragma unroll
    for (int b = 0; b < BB; ++b) { m[b] = -INFINITY; sum[b] = 0.f; }

    issue_tile(logits, s, v0, &tile[0][0][0]);
    for (int k = 0; k < NT; ++k) {
        if (k + 1 < NT) {
            issue_tile(logits, s, v0 + (k + 1) * TT, &tile[(k + 1) & 1][0][0]);
            wait_async_le16();                         // tile k done (in-order)
        } else {
            wait_async_0();
        }
        __syncthreads();                               // all waves' tiles visible
        const float* buf = &tile[k & 1][0][0];
        const int vb = v0 + k * TT;
#pragma unroll
        for (int u = 0; u < TT / 256; ++u) {
            int vl = t + u * 256;
            int v  = vb + vl;
            if (v < VOC) {
                float x[BB]; bool allneg = true;
#pragma unroll
                for (int b = 0; b < BB; ++b) { x[b] = buf[b * TT + vl]; allneg &= (x[b] < 0.f); }
                unsigned bit = (mw[(v - v0) >> 5] >> (v & 31)) & 1u;
#pragma unroll
                for (int b = 0; b < BB; ++b) {
                    float z = x[b];
                    if (bit) z = allneg ? z * PEN : z * INV_PEN;
                    z *= INV_TEMP;
                    float nm = fmaxf(m[b], z);
                    sum[b] = sum[b] * exp2f((m[b] - nm) * LOG2E) + exp2f((z - nm) * LOG2E);
                    m[b] = nm;
                }
            }
        }
        __syncthreads();                               // safe buffer reuse
    }

    // wave32 reduction, then cross-wave via LDS
#pragma unroll
    for (int b = 0; b < BB; ++b)
        for (int off = 16; off > 0; off >>= 1) {
            float om = __shfl_xor(m[b],   off, 32);
            float os = __shfl_xor(sum[b], off, 32);
            combine(m[b], sum[b], om, os);
        }
    int w = t >> 5;
    if ((t & 31) == 0) {
#pragma unroll
        for (int b = 0; b < BB; ++b) { redm[w][b] = m[b]; reds[w][b] = sum[b]; }
    }
    __syncthreads();
    if (t < BB) {
        float fm = -INFINITY, fs = 0.f;
#pragma unroll
        for (int ww = 0; ww < 8; ++ww) combine(fm, fs, redm[ww][t], reds[ww][t]);
        int o = ((s * NCH + c) * BB + t) * 2;
        part[o] = fm; part[o + 1] = fs;
    }
}

// ---- kernel 4: combine chunk partials -> final (max,sum) per (s,b) ----
__global__ void k_combine(const float* __restrict__ part, float* __restrict__ fin) {
    int t = threadIdx.x;                               // 512 threads: s*8+b
    int s = t >> 3, b = t & 7;
    float m = -INFINITY, sum = 0.f;
    for (int c = 0; c < NCH; ++c) {
        int o = ((s * NCH + c) * BB + b) * 2;
        combine(m, sum, part[o], part[o + 1]);
    }
    fin[(s * BB + b) * 2] = m; fin[(s * BB + b) * 2 + 1] = sum;
}

// ---- kernel 5: normalize pass ----
__global__ __launch_bounds__(256) void k_norm(const float* __restrict__ logits,
                                              const unsigned* __restrict__ mask,
                                              const float* __restrict__ fin,
                                              float* __restrict__ out) {
    __shared__ __align__(16) float tile[2][BB][TT];
    __shared__ unsigned mw[CHV / 32];
    __shared__ float fm[BB], fs[BB];
    const int t  = threadIdx.x;
    const int s  = blockIdx.x;
    const int c  = blockIdx.y;
    const int v0 = c * CHV;

    if (t < BB) { fm[t] = fin[(s * BB + t) * 2]; fs[t] = fin[(s * BB + t) * 2 + 1]; }
    if (t < CHV / 32) {
        int wi = v0 / 32 + t;
        mw[t] = (wi < VW) ? mask[s * VW + wi] : 0u;
    }

    float rm[BB], ri[BB];
    issue_tile(logits, s, v0, &tile[0][0][0]);
    for (int k = 0; k < NT; ++k) {
        if (k + 1 < NT) {
            issue_tile(logits, s, v0 + (k + 1) * TT, &tile[(k + 1) & 1][0][0]);
            wait_async_le16();
        } else {
            wait_async_0();
        }
        __syncthreads();
        if (k == 0) {
#pragma unroll
            for (int b = 0; b < BB; ++b) { rm[b] = fm[b]; ri[b] = 1.0f / fs[b]; }
        }
        const float* buf = &tile[k & 1][0][0];
        const int vb = v0 + k * TT;
#pragma unroll
        for (int u = 0; u < TT / 256; ++u) {
            int vl = t + u * 256;
            int v  = vb + vl;
            if (v < VOC) {
                float x[BB]; bool allneg = true;
#pragma unroll
                for (int b = 0; b < BB; ++b) { x[b] = buf[b * TT + vl]; allneg &= (x[b] < 0.f); }
                unsigned bit = (mw[(v - v0) >> 5] >> (v & 31)) & 1u;
#pragma unroll
                for (int b = 0; b < BB; ++b) {
                    float z = x[b];
                    if (bit) z = allneg ? z * PEN : z * INV_PEN;
                    z *= INV_TEMP;
                    out[((size_t)b * SS + s) * (size_t)VOC + v] =
                        exp2f((z - rm[b]) * LOG2E) * ri[b];
                }
            }
        }
        __syncthreads();
    }
}

// ---- workspace layout ----
#define WS_MASK_BYTES  ((size_t)(SS * VW) * 4u)            // 402,176
#define WS_PART_OFF    ((WS_MASK_BYTES + 511u) & ~511u)    // 402,432
#define WS_PART_BYTES  ((size_t)(SS * NCH * BB * 2) * 4u)  // 102,400
#define WS_FIN_OFF     (WS_PART_OFF + WS_PART_BYTES)

extern "C" void kernel_launch(void* const* d_in, const int* in_sizes, int n_in,
                              void* d_out, int out_size, void* d_ws, size_t ws_size,
                              hipStream_t stream) {
    const float* logits = (const float*)d_in[0];
    const int*   prev   = (const int*)d_in[1];
    float*       out    = (float*)d_out;
    unsigned char* ws   = (unsigned char*)d_ws;

    unsigned* mask = (unsigned*)ws;
    float*    part = (float*)(ws + WS_PART_OFF);
    float*    fin  = (float*)(ws + WS_FIN_OFF);
    int P = in_sizes[1] / SS;                              // 200

    k_maskzero<<<(SS * VW + 255) / 256, 256, 0, stream>>>(mask);
    k_maskfill<<<(SS * P + 255) / 256, 256, 0, stream>>>(prev, mask, P);

    dim3 grid(SS, NCH);                                    // 64 x 25 = 1600 blocks
    k_reduce<<<grid, 256, 0, stream>>>(logits, mask, part);
    k_combine<<<1, 512, 0, stream>>>(part, fin);
    k_norm<<<grid, 256, 0, stream>>>(logits, mask, fin, out);
}